// star_gcn_10746008175460
// MI455X (gfx1250) — compile-verified
//
#include <hip/hip_runtime.h>

#define NUM_USER 100000
#define NUM_ITEM 50000
#define N_NODES  (NUM_USER + NUM_ITEM)
#define N_EDGES  2400000
#define GDIM     64
#define SLOPE    0.1f

typedef float v2f __attribute__((ext_vector_type(2)));
typedef float v8f __attribute__((ext_vector_type(8)));

__device__ __forceinline__ float lrelu(float x) { return x >= 0.0f ? x : SLOPE * x; }

// ---------------------------------------------------------------------------
// x0 = concat(emb_user, emb_item)   (float4 copies)
// ---------------------------------------------------------------------------
__global__ void concat_kernel(const float* __restrict__ u,
                              const float* __restrict__ it,
                              float* __restrict__ out) {
    size_t i = (size_t)blockIdx.x * blockDim.x + threadIdx.x;   // float4 units
    const size_t nu4 = (size_t)NUM_USER * GDIM / 4;
    const size_t nt4 = (size_t)N_NODES  * GDIM / 4;
    if (i >= nt4) return;
    float4 v = (i < nu4) ? ((const float4*)u)[i] : ((const float4*)it)[i - nu4];
    ((float4*)out)[i] = v;
}

__global__ void zero_kernel(float* __restrict__ p, size_t n4) {
    size_t i = (size_t)blockIdx.x * blockDim.x + threadIdx.x;
    if (i < n4) ((float4*)p)[i] = make_float4(0.f, 0.f, 0.f, 0.f);
}

// ---------------------------------------------------------------------------
// SpMM scatter: out[row[e], :] += val[e] * x[col[e], :]
// 16 threads per edge, each thread handles 4 dims (float4 gather + 4 f32 atomics).
// x table (38.4 MB) stays L2-resident; atomics resolve at L2.
// ---------------------------------------------------------------------------
__global__ void spmm_scatter(const int* __restrict__ row,
                             const int* __restrict__ col,
                             const float* __restrict__ val,
                             const float* __restrict__ x,
                             float* __restrict__ out) {
    size_t t = (size_t)blockIdx.x * blockDim.x + threadIdx.x;
    size_t e = t >> 4;
    int    q = (int)(t & 15);
    if (e >= (size_t)N_EDGES) return;
    int   r = row[e];
    int   c = col[e];
    float v = val[e];
    float4 xv = ((const float4*)x)[(size_t)c * 16 + q];
    float* o  = out + (size_t)r * GDIM + (size_t)q * 4;
    atomicAdd(o + 0, v * xv.x);
    atomicAdd(o + 1, v * xv.y);
    atomicAdd(o + 2, v * xv.z);
    atomicAdd(o + 3, v * xv.w);
}

// ---------------------------------------------------------------------------
// Fused dense chain per 16-row stripe (one wave each):
//   a  = leaky(S)          (staged to LDS tile)
//   h  = a @ W1^T          (WMMA f32 16x16x4, tiles -> LDS)
//   g  = leaky(h @ W3^T)   (WMMA, tiles -> LDS)
//   xn = g @ W4^T          (WMMA, direct global store)
// ---------------------------------------------------------------------------
__device__ __forceinline__ void gemm_tile16x64(const float* __restrict__ tile,  // 16x64 A in LDS
                                               const float* __restrict__ W,     // 64x64 in LDS
                                               int lane, v8f acc[4]) {
    const int half = lane >> 4;        // 0 or 1
    const int mr   = lane & 15;
    const v8f z = {0.f,0.f,0.f,0.f,0.f,0.f,0.f,0.f};
#pragma unroll
    for (int j = 0; j < 4; ++j) acc[j] = z;
#pragma unroll
    for (int k = 0; k < GDIM; k += 4) {
        const int kb = k + 2 * half;
        v2f a = *(const v2f*)(tile + mr * GDIM + kb);
#pragma unroll
        for (int j = 0; j < 4; ++j) {
            v2f b = *(const v2f*)(W + (j * 16 + mr) * GDIM + kb);
            acc[j] = __builtin_amdgcn_wmma_f32_16x16x4_f32(
                /*neg_a=*/false, a, /*neg_b=*/false, b,
                /*c_mod=*/(short)0, acc[j], /*reuse_a=*/false, /*reuse_b=*/false);
        }
    }
}

__device__ __forceinline__ void tiles_to_lds(float* __restrict__ tile, int lane,
                                             const v8f acc[4], bool act) {
    const int half = lane >> 4;
    const int nc   = lane & 15;
#pragma unroll
    for (int j = 0; j < 4; ++j) {
#pragma unroll
        for (int r = 0; r < 8; ++r) {
            float v = acc[j][r];
            if (act) v = lrelu(v);
            tile[(half * 8 + r) * GDIM + j * 16 + nc] = v;
        }
    }
}

__global__ __launch_bounds__(256)
void dense_chain(const float* __restrict__ S,
                 const float* __restrict__ W1,
                 const float* __restrict__ W3,
                 const float* __restrict__ W4,
                 float* __restrict__ out, int nrows) {
    __shared__ float sW1[GDIM * GDIM];
    __shared__ float sW3[GDIM * GDIM];
    __shared__ float sW4[GDIM * GDIM];
    __shared__ float sT[8][16 * GDIM];   // per-wave 16x64 staging tile

    const int tid = threadIdx.x;
    for (int i = tid; i < GDIM * GDIM; i += 256) {
        sW1[i] = W1[i];
        sW3[i] = W3[i];
        sW4[i] = W4[i];
    }
    __syncthreads();

    const int wave = tid >> 5;
    const int lane = tid & 31;
    const int m0   = (blockIdx.x * 8 + wave) * 16;
    if (m0 >= nrows) return;            // whole-wave uniform; EXEC stays all-1s

    float* tile = sT[wave];

    // Stage A = leaky(S[m0:m0+16, :]) — contiguous 4 KB chunk, float4 coalesced.
    const float4* src4 = (const float4*)(S + (size_t)m0 * GDIM);
#pragma unroll
    for (int i = lane; i < 16 * GDIM / 4; i += 32) {
        float4 v = src4[i];
        v.x = lrelu(v.x); v.y = lrelu(v.y); v.z = lrelu(v.z); v.w = lrelu(v.w);
        ((float4*)tile)[i] = v;
    }

    v8f acc[4];
    // GEMM 1: h = a @ W1^T  (no activation)
    gemm_tile16x64(tile, sW1, lane, acc);
    tiles_to_lds(tile, lane, acc, /*act=*/false);
    // GEMM 2: g = leaky(h @ W3^T)
    gemm_tile16x64(tile, sW3, lane, acc);
    tiles_to_lds(tile, lane, acc, /*act=*/true);
    // GEMM 3: x_next = g @ W4^T  -> global
    gemm_tile16x64(tile, sW4, lane, acc);

    const int half = lane >> 4;
    const int nc   = lane & 15;
    float* dst = out + (size_t)m0 * GDIM;
#pragma unroll
    for (int j = 0; j < 4; ++j) {
#pragma unroll
        for (int r = 0; r < 8; ++r) {
            dst[(size_t)(half * 8 + r) * GDIM + j * 16 + nc] = acc[j][r];
        }
    }
}

// ---------------------------------------------------------------------------
extern "C" void kernel_launch(void* const* d_in, const int* in_sizes, int n_in,
                              void* d_out, int out_size, void* d_ws, size_t ws_size,
                              hipStream_t stream) {
    const int*   edge_row = (const int*)d_in[0];
    const int*   edge_col = (const int*)d_in[1];
    const float* edge_val = (const float*)d_in[2];
    const float* emb_user = (const float*)d_in[3];
    const float* emb_item = (const float*)d_in[4];
    const float* W_h1     = (const float*)d_in[5];
    const float* W_3      = (const float*)d_in[6];
    const float* W_4      = (const float*)d_in[7];

    float* out = (float*)d_out;
    float* x0  = out;                               // [N, 64]
    float* x1  = out + (size_t)N_NODES * GDIM;      // [N, 64]
    float* x2  = out + (size_t)2 * N_NODES * GDIM;  // [N, 64]
    float* S   = (float*)d_ws;                      // SpMM accumulator [N, 64]

    const size_t nt4   = (size_t)N_NODES * GDIM / 4;
    const int    cpBlk = 256;
    const int    cpGrd = (int)((nt4 + cpBlk - 1) / cpBlk);

    // x0 = concat(emb_user, emb_item)
    concat_kernel<<<cpGrd, cpBlk, 0, stream>>>(emb_user, emb_item, x0);

    const size_t spThreads = (size_t)N_EDGES * 16;
    const int    spGrd     = (int)((spThreads + 255) / 256);
    const int    tiles     = N_NODES / 16;          // 9375
    const int    dcGrd     = (tiles + 7) / 8;

    // ---- layer 1: x0 -> x1 ----
    zero_kernel<<<cpGrd, cpBlk, 0, stream>>>(S, nt4);
    spmm_scatter<<<spGrd, 256, 0, stream>>>(edge_row, edge_col, edge_val, x0, S);
    dense_chain<<<dcGrd, 256, 0, stream>>>(S, W_h1, W_3, W_4, x1, N_NODES);

    // ---- layer 2: x1 -> x2 ----
    zero_kernel<<<cpGrd, cpBlk, 0, stream>>>(S, nt4);
    spmm_scatter<<<spGrd, 256, 0, stream>>>(edge_row, edge_col, edge_val, x1, S);
    dense_chain<<<dcGrd, 256, 0, stream>>>(S, W_h1, W_3, W_4, x2, N_NODES);
}